// BrainNetGAT_88742614270025
// MI455X (gfx1250) — compile-verified
//
#include <hip/hip_runtime.h>

#define R 116
#define G 128
#define NNODE (R * G)     // 14848
#define HID 64
#define KENC 132          // IN_CH(116) + EMB(16)

typedef float v2f __attribute__((ext_vector_type(2)));
typedef float v8f __attribute__((ext_vector_type(8)));

// One wave computes a 16x16 f32 tile: acc += A[16 x K] * B[K x 16], A/B row-major in LDS.
// Layouts per CDNA5 ISA 7.12.2:
//  A 16x4: lanes 0-15 -> M=lane, VGPR0=K0,VGPR1=K1 ; lanes 16-31 -> K2,K3
//  B 4x16: lanes hold N, VGPR0/1 hold K rows (mirrored)
//  C/D:    VGPR i -> row (i + 8*(lane/16)), col = lane%16
__device__ __forceinline__ v8f wmma_k(const float* __restrict__ A, int lda,
                                      const float* __restrict__ B, int ldb,
                                      int K, v8f acc, int grp, int mn) {
  for (int k = 0; k < K; k += 4) {
    v2f a, b;
    a.x = A[mn * lda + k + 2 * grp];
    a.y = A[mn * lda + k + 2 * grp + 1];
    b.x = B[(k + 2 * grp) * ldb + mn];
    b.y = B[(k + 2 * grp + 1) * ldb + mn];
    acc = __builtin_amdgcn_wmma_f32_16x16x4_f32(false, a, false, b, (short)0, acc,
                                                false, false);
  }
  return acc;
}

// ---------------- Encoder: h = relu([x | emb[gid]] @ enc_W + enc_b) ----------------
__global__ __launch_bounds__(256) void k_encode(const float* __restrict__ x,
                                                const float* __restrict__ emb,
                                                const int* __restrict__ gid,
                                                const float* __restrict__ encW,
                                                const float* __restrict__ encb,
                                                float* __restrict__ h) {
  __shared__ float As[32 * KENC];
  __shared__ float Bs[KENC * 64];
  const int tid = threadIdx.x;
  const int r0 = blockIdx.x * 32;
  for (int i = tid; i < KENC * 64; i += 256) Bs[i] = encW[i];
  for (int i = tid; i < 32 * KENC; i += 256) {
    int row = i / KENC, k = i % KENC;
    int r = r0 + row;
    As[i] = (k < 116) ? x[(size_t)r * 116 + k] : emb[gid[r] * 16 + (k - 116)];
  }
  __syncthreads();
  const int lane = tid & 31, wave = tid >> 5;
  const int grp = lane >> 4, mn = lane & 15;
  const int mt = wave >> 2, nt = wave & 3;   // 2 m-tiles x 4 n-tiles = 8 waves
  v8f acc = {};
  acc = wmma_k(As + mt * 16 * KENC, KENC, Bs + nt * 16, 64, KENC, acc, grp, mn);
  int col = nt * 16 + mn;
  float bv = encb[col];
  #pragma unroll
  for (int i = 0; i < 8; ++i) {
    int row = r0 + mt * 16 + i + 8 * grp;
    h[(size_t)row * 64 + col] = fmaxf(acc[i] + bv, 0.f);
  }
}

// ---------------- BatchNorm batch stats (mu, var) over N per channel ----------------
__global__ __launch_bounds__(256) void k_bnstats(const float* __restrict__ h,
                                                 float* __restrict__ stats) {
  __shared__ float s1[256], s2[256];
  const int c = blockIdx.x, tid = threadIdx.x;
  float a1 = 0.f, a2 = 0.f;
  for (int r = tid; r < NNODE; r += 256) {
    float v = h[(size_t)r * 64 + c];
    a1 += v; a2 += v * v;
  }
  s1[tid] = a1; s2[tid] = a2;
  __syncthreads();
  for (int s = 128; s > 0; s >>= 1) {
    if (tid < s) { s1[tid] += s1[tid + s]; s2[tid] += s2[tid + s]; }
    __syncthreads();
  }
  if (tid == 0) {
    float mu = s1[0] / (float)NNODE;
    stats[c] = mu;
    stats[64 + c] = s2[0] / (float)NNODE - mu * mu;
  }
}

// ---------------- GINEConv (BN fused on load) + MLP, per graph ----------------
__global__ __launch_bounds__(256) void k_gine(float* __restrict__ h,
                                              const float* __restrict__ stats,
                                              const float* __restrict__ bn_g,
                                              const float* __restrict__ bn_b,
                                              const float* __restrict__ edge_attr,
                                              const float* __restrict__ We,
                                              const float* __restrict__ be,
                                              const float* __restrict__ W1,
                                              const float* __restrict__ b1,
                                              const float* __restrict__ W2,
                                              const float* __restrict__ b2) {
  __shared__ float A[128 * 64];   // hn, then t0 = hn+agg
  __shared__ float Bb[128 * 64];  // agg, then t1
  __shared__ float Ws[64 * 64];
  __shared__ float ea[R * 5];
  __shared__ float wes[5 * 64], bes[64];
  const int tid = threadIdx.x, g = blockIdx.x;
  for (int i = tid; i < 5 * 64; i += 256) wes[i] = We[i];
  if (tid < 64) bes[tid] = be[tid];
  for (int i = tid; i < 128 * 64; i += 256) {
    int r = i >> 6, c = i & 63;
    float v = 0.f;
    if (r < R) {
      float hv = h[((size_t)g * R + r) * 64 + c];
      v = (hv - stats[c]) * rsqrtf(stats[64 + c] + 1e-5f) * bn_g[c] + bn_b[c];
    }
    A[i] = v;
    Bb[i] = 0.f;
  }
  __syncthreads();
  for (int s = 0; s < R; ++s) {  // agg[d] += relu(hn[s] + lin_edge(e_sd))
    const float* base = edge_attr + (size_t)((g * R + s) * R) * 5;
    for (int i = tid; i < R * 5; i += 256) ea[i] = base[i];
    __syncthreads();
    for (int i = tid; i < R * 64; i += 256) {
      int d = i >> 6, c = i & 63;
      float e = bes[c];
      #pragma unroll
      for (int j = 0; j < 5; ++j) e += ea[d * 5 + j] * wes[j * 64 + c];
      Bb[d * 64 + c] += fmaxf(A[s * 64 + c] + e, 0.f);
    }
    __syncthreads();
  }
  for (int i = tid; i < 128 * 64; i += 256) A[i] += Bb[i];  // t0
  for (int i = tid; i < 64 * 64; i += 256) Ws[i] = W1[i];
  __syncthreads();
  const int lane = tid & 31, wave = tid >> 5;
  const int grp = lane >> 4, mn = lane & 15;
  for (int t = wave; t < 32; t += 8) {  // t1 = relu(t0 @ W1 + b1)
    int mt = t >> 2, nt = t & 3;
    v8f acc = {};
    acc = wmma_k(A + mt * 16 * 64, 64, Ws + nt * 16, 64, 64, acc, grp, mn);
    int col = nt * 16 + mn;
    float bv = b1[col];
    #pragma unroll
    for (int i = 0; i < 8; ++i)
      Bb[(mt * 16 + i + 8 * grp) * 64 + col] = fmaxf(acc[i] + bv, 0.f);
  }
  __syncthreads();
  for (int i = tid; i < 64 * 64; i += 256) Ws[i] = W2[i];
  __syncthreads();
  for (int t = wave; t < 32; t += 8) {  // h = relu(t1 @ W2 + b2)
    int mt = t >> 2, nt = t & 3;
    v8f acc = {};
    acc = wmma_k(Bb + mt * 16 * 64, 64, Ws + nt * 16, 64, 64, acc, grp, mn);
    int col = nt * 16 + mn;
    float bv = b2[col];
    #pragma unroll
    for (int i = 0; i < 8; ++i) {
      int row = mt * 16 + i + 8 * grp;
      if (row < R) h[((size_t)g * R + row) * 64 + col] = fmaxf(acc[i] + bv, 0.f);
    }
  }
}

// ---------------- GATv2 layer, per graph (dense 116x116 attention) ----------------
__global__ __launch_bounds__(256) void k_gat(float* __restrict__ h,
                                             const float* __restrict__ Wl,
                                             const float* __restrict__ bl,
                                             const float* __restrict__ Wr,
                                             const float* __restrict__ br,
                                             const float* __restrict__ att,
                                             const float* __restrict__ We,
                                             const float* __restrict__ bias,
                                             const float* __restrict__ edge_attr) {
  __shared__ float P[32768];  // [xl 8K][xr 8K][sc 16K (aliases: hn 8K + Ws 4K)]
  __shared__ float atts[64], wes[5 * 64];
  float* xl = P;
  float* xr = P + 8192;
  float* sc = P + 16384;
  float* hn = P + 16384;  // alias, dead before sc is used
  float* Ws = P + 24576;  // alias, dead before sc is used
  const int tid = threadIdx.x, g = blockIdx.x;
  if (tid < 64) atts[tid] = att[tid];
  for (int i = tid; i < 5 * 64; i += 256) wes[i] = We[i];
  for (int i = tid; i < 128 * 64; i += 256) {
    int r = i >> 6, c = i & 63;
    hn[i] = (r < R) ? h[((size_t)g * R + r) * 64 + c] : 0.f;
  }
  for (int i = tid; i < 64 * 64; i += 256) Ws[i] = Wl[i];
  __syncthreads();
  const int lane = tid & 31, wave = tid >> 5;
  const int grp = lane >> 4, mn = lane & 15;
  for (int t = wave; t < 32; t += 8) {  // xl = hn @ Wl + bl
    int mt = t >> 2, nt = t & 3;
    v8f acc = {};
    acc = wmma_k(hn + mt * 16 * 64, 64, Ws + nt * 16, 64, 64, acc, grp, mn);
    int col = nt * 16 + mn;
    float bv = bl[col];
    #pragma unroll
    for (int i = 0; i < 8; ++i) xl[(mt * 16 + i + 8 * grp) * 64 + col] = acc[i] + bv;
  }
  __syncthreads();
  for (int i = tid; i < 64 * 64; i += 256) Ws[i] = Wr[i];
  __syncthreads();
  for (int t = wave; t < 32; t += 8) {  // xr = hn @ Wr + br
    int mt = t >> 2, nt = t & 3;
    v8f acc = {};
    acc = wmma_k(hn + mt * 16 * 64, 64, Ws + nt * 16, 64, 64, acc, grp, mn);
    int col = nt * 16 + mn;
    float bv = br[col];
    #pragma unroll
    for (int i = 0; i < 8; ++i) xr[(mt * 16 + i + 8 * grp) * 64 + col] = acc[i] + bv;
  }
  __syncthreads();
  for (int i = tid; i < 128 * 128; i += 256) sc[i] = 0.f;  // kills hn/Ws
  __syncthreads();
  // scores[d][s] = dot(leaky_relu(xl[s]+xr[d]+ea@We), att)
  for (int e = tid; e < R * R; e += 256) {
    int s = e / R, d = e % R;  // edge id = s*R + d
    const float* eap = edge_attr + (size_t)((g * R + s) * R + d) * 5;
    float a0 = eap[0], a1 = eap[1], a2 = eap[2], a3 = eap[3], a4 = eap[4];
    float sum = 0.f;
    for (int c = 0; c < 64; ++c) {
      float ew = a0 * wes[c] + a1 * wes[64 + c] + a2 * wes[128 + c] +
                 a3 * wes[192 + c] + a4 * wes[256 + c];
      float z = xl[s * 64 + c] + xr[d * 64 + c] + ew;
      z = (z > 0.f) ? z : 0.2f * z;
      sum += z * atts[c];
    }
    sc[d * 128 + s] = sum;
  }
  __syncthreads();
  if (tid < R) {  // softmax over s for each destination d (padded cols stay 0)
    float* row = sc + tid * 128;
    float m = -3.4e38f;
    for (int s = 0; s < R; ++s) m = fmaxf(m, row[s]);
    float sum = 0.f;
    for (int s = 0; s < R; ++s) { float ev = __expf(row[s] - m); row[s] = ev; sum += ev; }
    float inv = 1.f / (sum + 1e-16f);
    for (int s = 0; s < R; ++s) row[s] *= inv;
  }
  __syncthreads();
  for (int t = wave; t < 32; t += 8) {  // h = relu(alpha @ xl + bias) : [128x128]@[128x64]
    int mt = t >> 2, nt = t & 3;
    v8f acc = {};
    acc = wmma_k(sc + mt * 16 * 128, 128, xl + nt * 16, 64, 128, acc, grp, mn);
    int col = nt * 16 + mn;
    float bv = bias[col];
    #pragma unroll
    for (int i = 0; i < 8; ++i) {
      int row = mt * 16 + i + 8 * grp;
      if (row < R) h[((size_t)g * R + row) * 64 + col] = fmaxf(acc[i] + bv, 0.f);
    }
  }
}

// ---------------- Attention pooling: per-node score ----------------
__global__ __launch_bounds__(256) void k_poolscore(const float* __restrict__ h,
                                                   const float* __restrict__ W1,
                                                   const float* __restrict__ b1,
                                                   const float* __restrict__ w2,
                                                   float* __restrict__ pscore) {
  __shared__ float Ws[64 * 64];
  const int tid = threadIdx.x;
  for (int i = tid; i < 4096; i += 256) Ws[i] = W1[i];
  __syncthreads();
  int r = blockIdx.x * 256 + tid;
  if (r >= NNODE) return;
  float hr[64];
  #pragma unroll
  for (int k = 0; k < 64; ++k) hr[k] = h[(size_t)r * 64 + k];
  float s = 0.f;
  for (int c = 0; c < 64; ++c) {
    float v = b1[c];
    for (int k = 0; k < 64; ++k) v += hr[k] * Ws[k * 64 + c];
    s += tanhf(v) * w2[c];
  }
  pscore[r] = s;
}

// ---------------- Global softmax reduction (max, sum) ----------------
__global__ __launch_bounds__(256) void k_softmaxred(const float* __restrict__ pscore,
                                                    float* __restrict__ red) {
  __shared__ float sm[256];
  const int tid = threadIdx.x;
  float m = -3.4e38f;
  for (int r = tid; r < NNODE; r += 256) m = fmaxf(m, pscore[r]);
  sm[tid] = m;
  __syncthreads();
  for (int s = 128; s > 0; s >>= 1) {
    if (tid < s) sm[tid] = fmaxf(sm[tid], sm[tid + s]);
    __syncthreads();
  }
  float mx = sm[0];
  __syncthreads();
  float sum = 0.f;
  for (int r = tid; r < NNODE; r += 256) sum += __expf(pscore[r] - mx);
  sm[tid] = sum;
  __syncthreads();
  for (int s = 128; s > 0; s >>= 1) {
    if (tid < s) sm[tid] += sm[tid + s];
    __syncthreads();
  }
  if (tid == 0) { red[0] = mx; red[1] = sm[0]; }
}

// ---------------- Pool (scatter-add by graph) ----------------
__global__ __launch_bounds__(64) void k_pooled(const float* __restrict__ h,
                                               const float* __restrict__ pscore,
                                               const float* __restrict__ red,
                                               float* __restrict__ pooled) {
  const int g = blockIdx.x, c = threadIdx.x;
  const float mx = red[0], inv = 1.f / red[1];
  float acc = 0.f;
  for (int r = 0; r < R; ++r) {
    float w = __expf(pscore[g * R + r] - mx) * inv;
    acc += h[((size_t)g * R + r) * 64 + c] * w;
  }
  pooled[g * 64 + c] = acc;
}

// ---------------- Head: relu(pooled@lin1 + b1) @ lin2 + b2 ----------------
__global__ __launch_bounds__(128) void k_head(const float* __restrict__ pooled,
                                              const float* __restrict__ W1,
                                              const float* __restrict__ b1,
                                              const float* __restrict__ W2,
                                              const float* __restrict__ b2,
                                              float* __restrict__ out) {
  __shared__ float t[116];
  __shared__ float p[64];
  const int g = blockIdx.x, tid = threadIdx.x;
  if (tid < 64) p[tid] = pooled[g * 64 + tid];
  __syncthreads();
  if (tid < 116) {
    float v = b1[tid];
    for (int k = 0; k < 64; ++k) v += p[k] * W1[k * 116 + tid];
    t[tid] = fmaxf(v, 0.f);
  }
  __syncthreads();
  if (tid < 2) {
    float v = b2[tid];
    for (int j = 0; j < 116; ++j) v += t[j] * W2[j * 2 + tid];
    out[g * 2 + tid] = v;
  }
}

extern "C" void kernel_launch(void* const* d_in, const int* in_sizes, int n_in,
                              void* d_out, int out_size, void* d_ws, size_t ws_size,
                              hipStream_t stream) {
  const float* x         = (const float*)d_in[0];
  const float* edge_attr = (const float*)d_in[1];
  const float* emb       = (const float*)d_in[2];
  const float* enc_W     = (const float*)d_in[3];
  const float* enc_b     = (const float*)d_in[4];
  const float* bn_g      = (const float*)d_in[5];
  const float* bn_b      = (const float*)d_in[6];
  const float* gine_We   = (const float*)d_in[7];
  const float* gine_be   = (const float*)d_in[8];
  const float* gine_W1   = (const float*)d_in[9];
  const float* gine_b1   = (const float*)d_in[10];
  const float* gine_W2   = (const float*)d_in[11];
  const float* gine_b2   = (const float*)d_in[12];
  const float* gat_Wl    = (const float*)d_in[13];
  const float* gat_bl    = (const float*)d_in[14];
  const float* gat_Wr    = (const float*)d_in[15];
  const float* gat_br    = (const float*)d_in[16];
  const float* gat_att   = (const float*)d_in[17];
  const float* gat_We    = (const float*)d_in[18];
  const float* gat_bias  = (const float*)d_in[19];
  const float* pool_W1   = (const float*)d_in[20];
  const float* pool_b1   = (const float*)d_in[21];
  const float* pool_w2   = (const float*)d_in[22];
  const float* lin1_W    = (const float*)d_in[23];
  const float* lin1_b    = (const float*)d_in[24];
  const float* lin2_W    = (const float*)d_in[25];
  const float* lin2_b    = (const float*)d_in[26];
  const int*   group_ids = (const int*)d_in[29];
  float* out = (float*)d_out;

  // Workspace layout (floats): h[N*64] | stats[128] | pscore[N] | red[2] | pooled[G*64]
  float* ws     = (float*)d_ws;
  float* h      = ws;
  float* stats  = h + (size_t)NNODE * 64;
  float* pscore = stats + 128;
  float* red    = pscore + NNODE;
  float* pooled = red + 2;

  k_encode<<<NNODE / 32, 256, 0, stream>>>(x, emb, group_ids, enc_W, enc_b, h);
  k_bnstats<<<64, 256, 0, stream>>>(h, stats);
  k_gine<<<G, 256, 0, stream>>>(h, stats, bn_g, bn_b, edge_attr, gine_We, gine_be,
                                gine_W1, gine_b1, gine_W2, gine_b2);
  for (int l = 0; l < 2; ++l)
    k_gat<<<G, 256, 0, stream>>>(h, gat_Wl + l * 4096, gat_bl + l * 64,
                                 gat_Wr + l * 4096, gat_br + l * 64,
                                 gat_att + l * 64, gat_We + l * 320,
                                 gat_bias + l * 64, edge_attr);
  k_poolscore<<<(NNODE + 255) / 256, 256, 0, stream>>>(h, pool_W1, pool_b1, pool_w2, pscore);
  k_softmaxred<<<1, 256, 0, stream>>>(pscore, red);
  k_pooled<<<G, 64, 0, stream>>>(h, pscore, red, pooled);
  k_head<<<G, 128, 0, stream>>>(pooled, lin1_W, lin1_b, lin2_W, lin2_b, out);
}